// GFRM_2241972929124
// MI455X (gfx1250) — compile-verified
//
#include <hip/hip_runtime.h>

// -----------------------------------------------------------------------------
// out[b,u] = base[u] + bias[u] + sum_i (relu(m_ii*x_bi) - relu(m_ii)) * w[u,i,i]
// base[u]  = sum_k relu(matrix_flat[k]) * weight_flat[u,k],  k over D*D
//
// HBM-bound: weight stream is 620 MB -> ~27 us floor at 23.3 TB/s.
// Kernel 1: TDM (tensor_load_to_lds) double-buffers 37x256 weight tiles into
// padded LDS (pad_interval/pad_amount give the 260-float conflict-free row
// stride), waves accumulate base[] with V_WMMA_F32_16X16X4_F32 from LDS,
// synchronized with s_wait_tensorcnt. Kernel 2: deterministic reduction of
// per-block partials + tiny diagonal correction.
// -----------------------------------------------------------------------------

typedef float v2f __attribute__((ext_vector_type(2)));
typedef float v8f __attribute__((ext_vector_type(8)));

#define DD      2048
#define UU      37
#define BB      16
#define UPAD    48                 // 3 u-tiles of 16
#define KB      256                // k-chunk per block iteration
#define WSTRIDE 260                // padded LDS row stride (floats)
#define NBLK    1024               // blocks in base kernel
#define KTOT    (DD * DD)          // 4194304
#define NCHUNK  (KTOT / KB)        // 16384
#define NITER   (NCHUNK / NBLK)    // 16 chunks per block, exact

#if defined(__has_builtin)
#if __has_builtin(__builtin_amdgcn_tensor_load_to_lds) && \
    __has_builtin(__builtin_amdgcn_s_wait_tensorcnt)
#define HAVE_TDM 1
#endif
#endif
#ifndef HAVE_TDM
#define HAVE_TDM 0
#endif

#if HAVE_TDM
typedef unsigned int tdm_v4u __attribute__((ext_vector_type(4)));
typedef int          tdm_v4i __attribute__((ext_vector_type(4)));
typedef int          tdm_v8i __attribute__((ext_vector_type(8)));

// Build a 2-D TDM descriptor (D#) and issue TENSOR_LOAD_TO_LDS.
// Field packing per CDNA5 ISA.
__device__ __forceinline__ void tdm_load_2d_to_lds(
    const void* gsrc, unsigned lds_byte_off,
    unsigned tensor_d0, unsigned tensor_d1,
    unsigned tile_d0, unsigned tile_d1,
    unsigned d0_stride_elems,
    unsigned pad_en, unsigned pad_interval, unsigned pad_amount)
{
    unsigned long long ga = (unsigned long long)(uintptr_t)gsrc;
    tdm_v4u g0;
    g0[0] = 1u;                                   // count=1, user mode, no gather
    g0[1] = lds_byte_off;                         // bits 63:32  lds_addr
    g0[2] = (unsigned)ga;                         // bits 95:64  global_addr lo
    g0[3] = (unsigned)((ga >> 32) & 0x01FFFFFFu)  // bits 120:96 global_addr hi
          | (2u << 30);                           // bits 127:126 type=2 (image)
    tdm_v8i g1;
    g1[0] = (int)((2u << 16)                      // data_size = 4 bytes
          | (pad_en << 20)
          | (pad_interval << 22)                  // 7 -> pad after 256 dwords
          | (pad_amount << 25));                  // 3 -> pad 4 dwords
    g1[1] = (int)((tensor_d0 & 0xFFFFu) << 16);   // tensor_dim0[15:0] @ bits 63:48
    g1[2] = (int)((tensor_d0 >> 16)               // tensor_dim0[31:16]
          | ((tensor_d1 & 0xFFFFu) << 16));       // tensor_dim1[15:0]
    g1[3] = (int)(((tensor_d1 >> 16) & 0xFFFFu)   // tensor_dim1[31:16]
          | (tile_d0 << 16));                     // tile_dim0
    g1[4] = (int)(tile_d1 & 0xFFFFu);             // tile_dim1; tile_dim2 = 0
    g1[5] = (int)d0_stride_elems;                 // tensor_dim0_stride[31:0]
    g1[6] = 0;                                    // stride hi bits / dim1_stride
    g1[7] = 0;
    tdm_v4i z4 = {0, 0, 0, 0};
#if __clang_major__ >= 23
    tdm_v8i z8 = {0, 0, 0, 0, 0, 0, 0, 0};
    __builtin_amdgcn_tensor_load_to_lds(g0, g1, z4, z4, z8, 0);
#else
    __builtin_amdgcn_tensor_load_to_lds(g0, g1, z4, z4, 0);
#endif
}
#endif  // HAVE_TDM

// WMMA accumulation over one 256-k chunk staged in LDS.
// A (16x4, every row = relu(matrix) chunk): vgpr0 = K0/K2, vgpr1 = K1/K3.
// B (4x16, n = u within tile):              vgpr0 = rows k0/k1, vgpr1 = k2/k3.
__device__ __forceinline__ void wmma_chunk(const float* __restrict__ wf,
                                           const float* __restrict__ rf,
                                           int wave, int lane, v8f acc[3])
{
    const int kbase = wave * 32;            // each wave owns 32 k's of the chunk
    const int aoff  = (lane >> 4) << 1;     // 0 (lanes 0-15) or 2 (lanes 16-31)
    const int boff  = (lane >> 4);          // 0 or 1
    const int urow  = lane & 15;
    #pragma unroll
    for (int s = 0; s < 8; ++s) {
        const int k4 = kbase + s * 4;
        v2f a;
        a.x = fmaxf(rf[k4 + aoff], 0.f);    // relu at consume time
        a.y = fmaxf(rf[k4 + aoff + 1], 0.f);
        #pragma unroll
        for (int t = 0; t < 3; ++t) {
            v2f b;
            const float* wr = &wf[(t * 16 + urow) * WSTRIDE + k4];
            b.x = wr[boff];
            b.y = wr[2 + boff];
            acc[t] = __builtin_amdgcn_wmma_f32_16x16x4_f32(
                /*neg_a=*/false, a, /*neg_b=*/false, b,
                /*c_mod=*/(short)0, acc[t],
                /*reuse_a=*/false, /*reuse_b=*/false);
        }
    }
}

__global__ __launch_bounds__(256) void base_partial_kernel(
    const float* __restrict__ matrix,   // [D*D]
    const float* __restrict__ weight,   // [U][D*D]
    float* __restrict__ partial)        // [NBLK][UPAD]
{
#if HAVE_TDM
    __shared__ float w_lds[2][UPAD * WSTRIDE];  // double-buffered weight tiles
    __shared__ float r_lds[2][KB];              // double-buffered matrix chunks
#else
    __shared__ float w_lds[1][UPAD * WSTRIDE];
    __shared__ float r_lds[1][KB];
#endif
    __shared__ float red[8 * UPAD];

    const int tid  = threadIdx.x;
    const int wave = tid >> 5;
    const int lane = tid & 31;

    v8f acc[3];
    #pragma unroll
    for (int t = 0; t < 3; ++t) acc[t] = (v8f){0.f,0.f,0.f,0.f,0.f,0.f,0.f,0.f};

#if HAVE_TDM
    // Zero both weight buffers once: TDM fills rows 0..36 only; padded rows
    // 37..47 must read as 0 for the third (partial) u-tile.
    for (int i = tid; i < 2 * UPAD * WSTRIDE; i += 256)
        ((float*)w_lds)[i] = 0.f;
    __syncthreads();

    if (tid < 32) {  // wave 0 drives the TDM pipeline (EXEC-safe: branched around)
        const int k0 = blockIdx.x * KB;
        tdm_load_2d_to_lds(weight + k0, (unsigned)(uintptr_t)&w_lds[0][0],
                           KTOT, UU, KB, UU, KTOT, 1u, 7u, 3u);
        tdm_load_2d_to_lds(matrix + k0, (unsigned)(uintptr_t)&r_lds[0][0],
                           KTOT, 1u, KB, 1u, KTOT, 0u, 0u, 0u);
    }

    for (int it = 0; it < NITER; ++it) {
        const int cur = it & 1;
        if (tid < 32) {
            if (it + 1 < NITER) {
                // Prefetch next chunk into the spare buffer, then drain only
                // the current chunk's 2 TDM ops (TDM is in-order per wave).
                const int k0n = (blockIdx.x + (it + 1) * NBLK) * KB;
                tdm_load_2d_to_lds(weight + k0n,
                                   (unsigned)(uintptr_t)&w_lds[cur ^ 1][0],
                                   KTOT, UU, KB, UU, KTOT, 1u, 7u, 3u);
                tdm_load_2d_to_lds(matrix + k0n,
                                   (unsigned)(uintptr_t)&r_lds[cur ^ 1][0],
                                   KTOT, 1u, KB, 1u, KTOT, 0u, 0u, 0u);
                __builtin_amdgcn_s_wait_tensorcnt(2);
            } else {
                __builtin_amdgcn_s_wait_tensorcnt(0);
            }
        }
        __syncthreads();                       // tile visible to all waves
        wmma_chunk(&w_lds[cur][0], &r_lds[cur][0], wave, lane, acc);
        __syncthreads();                       // buffer free before refill
    }
#else
    // Fallback: synchronous staging (round-1 path), relu at consume time.
    for (int chunk = (int)blockIdx.x; chunk < NCHUNK; chunk += NBLK) {
        const int k0 = chunk * KB;
        __syncthreads();
        r_lds[0][tid] = matrix[k0 + tid];
        #pragma unroll
        for (int i = 0; i < 12; ++i) {
            int p    = tid + i * 256;          // 0..3071
            int row  = p >> 6;                 // 0..47
            int col4 = (p & 63) << 2;          // 0..252
            float4 v;
            if (row < UU)
                v = *(const float4*)(weight + (size_t)row * KTOT + k0 + col4);
            else
                v = make_float4(0.f, 0.f, 0.f, 0.f);
            *(float4*)(&w_lds[0][row * WSTRIDE + col4]) = v;
        }
        __syncthreads();
        wmma_chunk(&w_lds[0][0], &r_lds[0][0], wave, lane, acc);
    }
#endif

    // C row 0 lives in c[0], lanes 0-15 (N = lane); all rows identical.
    __syncthreads();
    if (lane < 16) {
        #pragma unroll
        for (int t = 0; t < 3; ++t)
            red[wave * UPAD + t * 16 + lane] = acc[t][0];
    }
    __syncthreads();
    if (tid < UPAD) {
        float s = 0.f;
        #pragma unroll
        for (int w = 0; w < 8; ++w) s += red[w * UPAD + tid];
        partial[blockIdx.x * UPAD + tid] = s;
    }
}

__global__ __launch_bounds__(256) void finalize_kernel(
    const float* __restrict__ inputs,   // [B][D]
    const float* __restrict__ matrix,   // [D][D]
    const float* __restrict__ weight,   // [U][D][D]
    const float* __restrict__ bias,     // [U]
    const float* __restrict__ partial,  // [NBLK][UPAD]
    float* __restrict__ out)            // [B][U]
{
    __shared__ float sred[256];
    const int u   = blockIdx.x;         // 0..36
    const int tid = threadIdx.x;

    // base[u]: deterministic tree reduction of NBLK partials
    float s = 0.f;
    for (int i = tid; i < NBLK; i += 256) s += partial[i * UPAD + u];
    sred[tid] = s;
    __syncthreads();
    for (int off = 128; off > 0; off >>= 1) {
        if (tid < off) sred[tid] += sred[tid + off];
        __syncthreads();
    }
    const float base_u = sred[0];
    __syncthreads();

    // Diagonal correction. Preload this block's diag slices (8 per thread).
    float wd[8], md[8];
    #pragma unroll
    for (int j = 0; j < 8; ++j) {
        const size_t i = (size_t)(tid + j * 256);
        md[j] = matrix[i * (DD + 1)];
        wd[j] = weight[(size_t)u * KTOT + i * (DD + 1)];
    }
    const float bu = bias[u];

    for (int b = 0; b < BB; ++b) {
        float c = 0.f;
        #pragma unroll
        for (int j = 0; j < 8; ++j) {
            const int i  = tid + j * 256;
            const float x  = inputs[b * DD + i];
            const float mx = md[j] * x;
            const float r1 = mx > 0.f ? mx : 0.f;
            const float r0 = md[j] > 0.f ? md[j] : 0.f;
            c += (r1 - r0) * wd[j];
        }
        sred[tid] = c;
        __syncthreads();
        for (int off = 128; off > 0; off >>= 1) {
            if (tid < off) sred[tid] += sred[tid + off];
            __syncthreads();
        }
        if (tid == 0) out[b * UU + u] = base_u + bu + sred[0];
        __syncthreads();
    }
}

extern "C" void kernel_launch(void* const* d_in, const int* in_sizes, int n_in,
                              void* d_out, int out_size, void* d_ws, size_t ws_size,
                              hipStream_t stream) {
    const float* inputs = (const float*)d_in[0];   // (16, 2048)
    const float* matrix = (const float*)d_in[1];   // (2048, 2048)
    const float* weight = (const float*)d_in[2];   // (37, 2048, 2048)
    const float* bias   = (const float*)d_in[3];   // (37,)
    float* out     = (float*)d_out;                // (16, 37)
    float* partial = (float*)d_ws;                 // NBLK*UPAD floats = 192 KB

    base_partial_kernel<<<NBLK, 256, 0, stream>>>(matrix, weight, partial);
    finalize_kernel<<<UU, 256, 0, stream>>>(inputs, matrix, weight, bias, partial, out);
}